// RegionLoss_83099027243120
// MI455X (gfx1250) — compile-verified
//
#include <hip/hip_runtime.h>
#include <math.h>

typedef __attribute__((ext_vector_type(2))) float v2f;
typedef __attribute__((ext_vector_type(4))) float v4f;
typedef __attribute__((ext_vector_type(8))) float v8f;

#define NB 128
#define NA 2
#define NH 128
#define NW 128
#define PLANE (NH * NW)
#define THRESH_F 0.6f
#define OBJ_SCALE_F 5.0f
#define NBLOCKS_B 4096  // 16 * NA * NB

// Workspace layout (as float indices into d_ws):
//  [0] coord loss sum (over batch), [1] obj-conf loss sum
//  [8   .. 136)  gt_x per b     [136 .. 264) gt_y
//  [264 .. 392)  gt_w           [392 .. 520) gt_h
//  ints (same 4B slots): [520..648) best_n, [648..776) gxi, [776..904) gyi
//  [1024 .. 5120)  per-block conf^2 partials
//  [5120 .. 9216)  per-block noobj-count partials
#define WS_COORD   0
#define WS_CONFOBJ 1
#define WS_GTX     8
#define WS_GTY     (8 + 128)
#define WS_GTW     (8 + 256)
#define WS_GTH     (8 + 384)
#define WS_BESTN   520
#define WS_GXI     648
#define WS_GYI     776
#define WS_C2PART  1024
#define WS_CNTPART (1024 + 4096)

// Full 32-lane fp32 sum via V_WMMA_F32_16X16X4_F32.
// A vgpr0 = v (=> A[m][0]=v_m from lanes 0..15, A[m][2]=v_{m+16}), A vgpr1 = 0,
// B = ones  =>  D[m][n] = v_m + v_{m+16} (broadcast over n).
// In-lane sum of the 8 D vgprs gives 8 of those rows; the opposite half-wave
// holds the other 8 -> one shfl_xor(16) completes the 32-lane total.
__device__ __forceinline__ float wave_reduce_f32(float v) {
  v2f a;
  a[0] = v;
  a[1] = 0.0f;
  v2f bones;
  bones[0] = 1.0f;
  bones[1] = 1.0f;
  v8f c = {};
  v8f d = __builtin_amdgcn_wmma_f32_16x16x4_f32(false, a, false, bones,
                                                (short)0, c, false, false);
  float s = d[0] + d[1] + d[2] + d[3] + d[4] + d[5] + d[6] + d[7];
  s += __shfl_xor(s, 16, 32);
  return s;  // total of all 32 lanes, valid in every lane
}

// sigmoid with hardware rcp (v_rcp_f32, ~1 ulp) instead of IEEE divide
__device__ __forceinline__ float sigmoidf_fast(float x) {
  return __builtin_amdgcn_rcpf(1.0f + __expf(-x));
}

// ---------------------------------------------------------------------------
// Kernel 1: per-batch prep + gather of selected cells + coord/obj-conf loss
// ---------------------------------------------------------------------------
__global__ __launch_bounds__(128) void region_prep_kernel(
    const float* __restrict__ out, const float* __restrict__ target,
    const float* __restrict__ anchors, float* __restrict__ wsf) {
  const int b = threadIdx.x;  // 128 threads, one per batch element
  const float t0 = target[b * 4 + 0];
  const float t1 = target[b * 4 + 1];
  const float t2 = target[b * 4 + 2];
  const float t3 = target[b * 4 + 3];
  const float gtx = t0 * (float)NW;
  const float gty = t1 * (float)NH;
  const float gtw = t2 * (float)NW;
  const float gth = t3 * (float)NH;

  const float a0w = anchors[0], a0h = anchors[1];
  const float a1w = anchors[2], a1h = anchors[3];
  const float garea = gtw * gth;
  const float i0 = fminf(gtw, a0w) * fminf(gth, a0h);
  const float r0 = i0 / (garea + 1e-16f + a0w * a0h - i0);
  const float i1 = fminf(gtw, a1w) * fminf(gth, a1h);
  const float r1 = i1 / (garea + 1e-16f + a1w * a1h - i1);
  const int best_n = (r1 > r0) ? 1 : 0;  // argmax, first-max on tie
  const float baw = best_n ? a1w : a0w;
  const float bah = best_n ? a1h : a0h;

  const int gxi = (int)gtx;  // positive -> trunc == floor
  const int gyi = (int)gty;
  const float tx = gtx - floorf(gtx);
  const float ty = gty - floorf(gty);
  const float tw = logf(gtw / baw + 1e-16f);
  const float th = logf(gth / bah + 1e-16f);
  const float scale = 2.0f - t2 * t3;

  const size_t base =
      (size_t)((b * NA + best_n) * 5) * PLANE + (size_t)(gyi * NW + gxi);
  const float o0 = out[base + 0 * PLANE];
  const float o1 = out[base + 1 * PLANE];
  const float o2 = out[base + 2 * PLANE];
  const float o3 = out[base + 3 * PLANE];
  const float o4 = out[base + 4 * PLANE];

  const float xs = sigmoidf_fast(o0);
  const float ys = sigmoidf_fast(o1);
  const float cs = sigmoidf_fast(o4);
  const float dx = scale * (xs - tx);
  const float dy = scale * (ys - ty);
  const float dw = scale * (o2 - tw);
  const float dh = scale * (o3 - th);
  float coord = dx * dx + dy * dy + dw * dw + dh * dh;
  float cobj = (cs - 1.0f) * (cs - 1.0f);

  // stash per-batch constants for the streaming pass
  wsf[WS_GTX + b] = gtx;
  wsf[WS_GTY + b] = gty;
  wsf[WS_GTW + b] = gtw;
  wsf[WS_GTH + b] = gth;
  int* wsi = (int*)wsf;
  wsi[WS_BESTN + b] = best_n;
  wsi[WS_GXI + b] = gxi;
  wsi[WS_GYI + b] = gyi;

  // reduce across the 4 waves of this block
  coord = wave_reduce_f32(coord);
  cobj = wave_reduce_f32(cobj);
  __shared__ float l1[4], l2[4];
  const int wv = b >> 5;
  if ((b & 31) == 0) {
    l1[wv] = coord;
    l2[wv] = cobj;
  }
  __syncthreads();
  if (b == 0) {
    float s1 = 0.f, s2 = 0.f;
#pragma unroll
    for (int w = 0; w < 4; ++w) {
      s1 += l1[w];
      s2 += l2[w];
    }
    wsf[WS_COORD] = s1;
    wsf[WS_CONFOBJ] = s2;
  }
}

// ---------------------------------------------------------------------------
// Kernel 2: streaming noobj pass. grid = (16, NA, NB), block = 256.
// Each thread: 4 consecutive pixels x 5 channels via non-temporal b128 loads.
// ---------------------------------------------------------------------------
__global__ __launch_bounds__(256) void region_noobj_kernel(
    const float* __restrict__ out, const float* __restrict__ anchors,
    const float* __restrict__ wsf, float* __restrict__ c2part,
    float* __restrict__ cntpart) {
  const int b = blockIdx.z;
  const int a = blockIdx.y;
  const int tid = threadIdx.x;
  const int pix0 = (blockIdx.x * 256 + tid) * 4;

  // block-uniform constants -> scalar loads
  const float aw = anchors[a * 2 + 0];
  const float ah = anchors[a * 2 + 1];
  const float gtx = wsf[WS_GTX + b];
  const float gty = wsf[WS_GTY + b];
  const float gtw = wsf[WS_GTW + b];
  const float gth = wsf[WS_GTH + b];
  const int* wsi = (const int*)wsf;
  const int best_n = wsi[WS_BESTN + b];
  const int gxi = wsi[WS_GXI + b];
  const int gyi = wsi[WS_GYI + b];
  const bool a_match = (a == best_n);

  const float gminx = gtx - 0.5f * gtw, gmaxx = gtx + 0.5f * gtw;
  const float gminy = gty - 0.5f * gth, gmaxy = gty + 0.5f * gth;
  const float garea = gtw * gth;

  const float* base = out + (size_t)((b * NA + a) * 5) * PLANE + pix0;
  // single-use stream: non-temporal 128-bit loads
  const v4f o0 = __builtin_nontemporal_load((const v4f*)(base + 0 * PLANE));
  const v4f o1 = __builtin_nontemporal_load((const v4f*)(base + 1 * PLANE));
  const v4f o2 = __builtin_nontemporal_load((const v4f*)(base + 2 * PLANE));
  const v4f o3 = __builtin_nontemporal_load((const v4f*)(base + 3 * PLANE));
  const v4f o4 = __builtin_nontemporal_load((const v4f*)(base + 4 * PLANE));

  const int gy = pix0 >> 7;         // nW == 128
  const int gxb = pix0 & (NW - 1);  // 4 pixels share one row
  const float gyf = (float)gy;

  float conf2 = 0.0f;
  float cnt = 0.0f;
#pragma unroll
  for (int j = 0; j < 4; ++j) {
    const float sx = sigmoidf_fast(o0[j]);
    const float sy = sigmoidf_fast(o1[j]);
    const float pw = __expf(o2[j]) * aw;
    const float ph = __expf(o3[j]) * ah;
    const float cf = sigmoidf_fast(o4[j]);
    const float px = sx + (float)(gxb + j);
    const float py = sy + gyf;
    const float hw = 0.5f * pw, hh = 0.5f * ph;
    const float mx = fminf(px - hw, gminx);
    const float Mx = fmaxf(px + hw, gmaxx);
    const float my = fminf(py - hh, gminy);
    const float My = fmaxf(py + hh, gmaxy);
    const float cw = pw + gtw - (Mx - mx);
    const float chh = ph + gth - (My - my);
    const float carea = cw * chh;
    const float uarea = pw * ph + garea - carea;
    // iou <= THRESH  <=>  carea <= THRESH*uarea (uarea > 0), no divide needed
    bool noobj = (cw <= 0.0f) || (chh <= 0.0f) || (carea <= THRESH_F * uarea);
    if (a_match && (gy == gyi) && ((gxb + j) == gxi)) noobj = false;
    if (noobj) {
      conf2 += cf * cf;
      cnt += 1.0f;
    }
  }

  // wave (WMMA) then block reduction; one partial pair per block
  conf2 = wave_reduce_f32(conf2);
  cnt = wave_reduce_f32(cnt);
  __shared__ float l1[8], l2[8];
  const int wv = tid >> 5;
  if ((tid & 31) == 0) {
    l1[wv] = conf2;
    l2[wv] = cnt;
  }
  __syncthreads();
  if (tid == 0) {
    float s1 = 0.f, s2 = 0.f;
#pragma unroll
    for (int w = 0; w < 8; ++w) {
      s1 += l1[w];
      s2 += l2[w];
    }
    const int bid = blockIdx.x + 16 * (blockIdx.y + NA * blockIdx.z);
    c2part[bid] = s1;
    cntpart[bid] = s2;
  }
}

// ---------------------------------------------------------------------------
// Kernel 3: final combine. 1 block x 256 threads over 4096 partial pairs.
// ---------------------------------------------------------------------------
__global__ __launch_bounds__(256) void region_final_kernel(
    const float* __restrict__ wsf, float* __restrict__ out_loss) {
  const int tid = threadIdx.x;
  float c2 = 0.0f, cn = 0.0f;
#pragma unroll
  for (int i = tid; i < NBLOCKS_B; i += 256) {
    c2 += wsf[WS_C2PART + i];
    cn += wsf[WS_CNTPART + i];
  }
  c2 = wave_reduce_f32(c2);
  cn = wave_reduce_f32(cn);
  __shared__ float l1[8], l2[8];
  const int wv = tid >> 5;
  if ((tid & 31) == 0) {
    l1[wv] = c2;
    l2[wv] = cn;
  }
  __syncthreads();
  if (tid == 0) {
    float s1 = 0.f, s2 = 0.f;
#pragma unroll
    for (int w = 0; w < 8; ++w) {
      s1 += l1[w];
      s2 += l2[w];
    }
    const float inv_nb = 1.0f / (float)NB;
    const float loss = wsf[WS_COORD] * inv_nb +
                       OBJ_SCALE_F * wsf[WS_CONFOBJ] * inv_nb + s1 / s2;
    out_loss[0] = loss;
  }
}

extern "C" void kernel_launch(void* const* d_in, const int* in_sizes, int n_in,
                              void* d_out, int out_size, void* d_ws,
                              size_t ws_size, hipStream_t stream) {
  (void)in_sizes;
  (void)n_in;
  (void)out_size;
  (void)ws_size;
  const float* output = (const float*)d_in[0];   // (128, 10, 128, 128) f32
  const float* target = (const float*)d_in[1];   // (128, 4) f32
  const float* anchors = (const float*)d_in[2];  // (2, 2) f32
  float* wsf = (float*)d_ws;
  float* loss = (float*)d_out;

  region_prep_kernel<<<1, 128, 0, stream>>>(output, target, anchors, wsf);
  dim3 grid(16, NA, NB);
  region_noobj_kernel<<<grid, 256, 0, stream>>>(
      output, anchors, wsf, wsf + WS_C2PART, wsf + WS_CNTPART);
  region_final_kernel<<<1, 256, 0, stream>>>(wsf, loss);
}